// GCN_83854941487765
// MI455X (gfx1250) — compile-verified
//
#include <hip/hip_runtime.h>
#include <hip/hip_bf16.h>
#include <math.h>

#define NN   50000
#define NE   800000
#define FIN  11
#define FPAD 16
#define HID  128
#define NCLS 16
#define NG   64
#define NEG_SLOPE 0.2f

typedef float        v2f __attribute__((ext_vector_type(2)));
typedef float        v8f __attribute__((ext_vector_type(8)));
typedef unsigned int v4u __attribute__((ext_vector_type(4)));
typedef int          v8i __attribute__((ext_vector_type(8)));
typedef int          v4i __attribute__((ext_vector_type(4)));

#if __has_builtin(__builtin_amdgcn_wmma_f32_16x16x4_f32)
#define HAVE_WMMA_F32 1
#endif
#if __has_builtin(__builtin_amdgcn_tensor_load_to_lds) && \
    __has_builtin(__builtin_amdgcn_s_wait_tensorcnt)
#define HAVE_TDM 1
#endif

#if defined(HAVE_TDM)
// Issue a 2D TDM load: tile (tdim0 x tdim1) elements of 4B from row-major
// tensor (stride0 dwords/row, valid extent xdim0 x xdim1 with OOB->0) into
// LDS at ldsAddr, padding padAmt dwords after every (1 << (padCode+1)) dwords.
// Toolchain uses the 6-arg builtin: (v4u, v8i, v4i, v4i, v8i, i32 cpol).
__device__ __forceinline__ void tdm_load_2d_f32(unsigned int ldsAddr,
                                                const float* gptr,
                                                unsigned int xdim0, unsigned int xdim1,
                                                unsigned int tdim0, unsigned int tdim1,
                                                unsigned int stride0,
                                                unsigned int padCode, unsigned int padAmtCode)
{
    const unsigned long long ga = (unsigned long long)(size_t)gptr;
    v4u g0;
    g0[0] = 1u;                                             // count=1, no gather
    g0[1] = ldsAddr;                                        // lds_addr (bytes)
    g0[2] = (unsigned int)(ga & 0xFFFFFFFFull);             // global_addr[31:0]
    g0[3] = (unsigned int)((ga >> 32) & 0x1FFFFFFull)       // global_addr[56:32]
          | (2u << 30);                                     // type = 2 ("image")
    v8i g1;
    g1[0] = (int)((2u << 16)                                // data_size = 4B
                | (1u << 20)                                // pad_enable
                | (padCode   << 22)                         // pad_interval
                | (padAmtCode << 25));                      // pad_amount
    g1[1] = (int)(xdim0 << 16);                             // tensor_dim0[15:0]
    g1[2] = (int)((xdim0 >> 16) | (xdim1 << 16));           // dim0 hi | dim1 lo
    g1[3] = (int)((xdim1 >> 16) | (tdim0 << 16));           // dim1 hi | tile_dim0
    g1[4] = (int)(tdim1 & 0xFFFFu);                         // tile_dim1 | tile_dim2=0
    g1[5] = (int)stride0;                                   // tensor_dim0_stride lo32
    g1[6] = 0;                                              // stride0 hi | stride1 lo
    g1[7] = 0;
    v4i g2 = {0, 0, 0, 0};
    v4i g3 = {0, 0, 0, 0};
    v8i gx = {0, 0, 0, 0, 0, 0, 0, 0};
    __builtin_amdgcn_tensor_load_to_lds(g0, g1, g2, g3, gx, 0);
}
#endif

// ---------------------------------------------------------------------------
// Branchless WMMA GEMM: H[nrows,COUT] = A[nrows,CIN] @ W[CIN,COUT]
// 256 threads = 8 waves; each wave owns one 16x16 tile (TILES_N col tiles x
// TILES_M row tiles). A-panel and full W staged in LDS (padded strides) via
// the Tensor Data Mover (fallback: cooperative float4 loads).
// CIN, COUT in {16,128}; GUARD=false when nrows % ROWS == 0.
// ---------------------------------------------------------------------------
template<int CIN, int COUT, bool GUARD>
__global__ __launch_bounds__(256)
void gat_gemm_wmma(const float* __restrict__ A, const float* __restrict__ W,
                   float* __restrict__ H, int nrows)
{
    constexpr int TILES_N = COUT / 16;          // 8 (COUT=128) or 1 (COUT=16)
    constexpr int TILES_M = 8 / TILES_N;        // 1 or 8
    constexpr int ROWS    = 16 * TILES_M;       // rows per block: 16 or 128
    constexpr int ASTR    = CIN + 4;            // padded LDS strides
    constexpr int WSTR    = COUT + 4;

    __shared__ float Asm[ROWS * ASTR];
    __shared__ float Wsm[CIN * WSTR];

    const int tid  = threadIdx.x;
    const int lane = tid & 31;
    const int wid  = tid >> 5;                  // wave id 0..7
    const int lr   = lane & 15;
    const int hs   = lane >> 4;                 // half-wave select
    const int nt   = wid % TILES_N;
    const int mt   = wid / TILES_N;
    const int rowBase = blockIdx.x * ROWS;

#if defined(HAVE_TDM)
    // pad_interval code: row length in dwords = 1 << (code+1); amount code 3 = 4 dwords
    constexpr unsigned int wPad = (COUT == 128) ? 6u : 3u;
    constexpr unsigned int aPad = (CIN  == 128) ? 6u : 3u;
    if (wid == 0) {
        tdm_load_2d_f32((unsigned int)(size_t)&Wsm[0], W,
                        COUT, CIN, COUT, CIN, COUT, wPad, 3u);
        __builtin_amdgcn_s_wait_tensorcnt(0);
    } else if (wid == 1) {
        const unsigned int remain = (unsigned int)(nrows - rowBase); // OOB rows -> 0
        tdm_load_2d_f32((unsigned int)(size_t)&Asm[0],
                        A + (size_t)rowBase * CIN,
                        CIN, remain, CIN, ROWS, CIN, aPad, 3u);
        __builtin_amdgcn_s_wait_tensorcnt(0);
    }
#else
    // Cooperative fill: W panel -> LDS (float4, COUT % 4 == 0)
    for (int idx = tid * 4; idx < CIN * COUT; idx += 256 * 4) {
        const int r = idx / COUT, c = idx % COUT;
        *(float4*)&Wsm[r * WSTR + c] = *(const float4*)(W + (size_t)r * COUT + c);
    }
    // Cooperative fill: A rows -> LDS (zero rows past nrows)
    for (int idx = tid * 4; idx < ROWS * CIN; idx += 256 * 4) {
        const int r = idx / CIN, c = idx % CIN;
        const int gr = rowBase + r;
        float4 v = make_float4(0.f, 0.f, 0.f, 0.f);
        if (gr < nrows) v = *(const float4*)(A + (size_t)gr * CIN + c);
        *(float4*)&Asm[r * ASTR + c] = v;
    }
#endif
    __syncthreads();

    v8f acc = {0.f, 0.f, 0.f, 0.f, 0.f, 0.f, 0.f, 0.f};

#if defined(HAVE_WMMA_F32)
    // A 16x4 f32 frag: lanes 0-15 -> K=0,1; lanes 16-31 -> K=2,3 (M = lane%16)
    // B 4x16 f32 frag: mirrored (N = lane%16)
    const float* aRow = &Asm[(mt * 16 + lr) * ASTR];
    #pragma unroll
    for (int k = 0; k < CIN; k += 4) {
        const int ka = k + 2 * hs;
        v2f a, b;
        a.x = aRow[ka];
        a.y = aRow[ka + 1];
        b.x = Wsm[(ka    ) * WSTR + nt * 16 + lr];
        b.y = Wsm[(ka + 1) * WSTR + nt * 16 + lr];
        acc = __builtin_amdgcn_wmma_f32_16x16x4_f32(false, a, false, b,
                                                    (short)0, acc, false, false);
    }
#endif
    // (host-compilation pass falls through with acc == 0; never executed)

    // C/D layout: VGPR r -> lanes 0-15: (M=r, N=lane); lanes 16-31: (M=8+r)
    const int row0 = rowBase + mt * 16 + hs * 8;
    const int col  = nt * 16 + lr;
    float* hp = H + (size_t)row0 * COUT + col;
    if (!GUARD || row0 + 7 < nrows) {
        #pragma unroll
        for (int r = 0; r < 8; ++r) hp[(size_t)r * COUT] = acc[r];
    } else {
        #pragma unroll
        for (int r = 0; r < 8; ++r)
            if (row0 + r < nrows) hp[(size_t)r * COUT] = acc[r];
    }
}

// ---------------------------------------------------------------------------
// Zero-pad x [N,11] -> xp [N,16]
// ---------------------------------------------------------------------------
__global__ void pack_x(const float* __restrict__ x, float* __restrict__ xp, int n)
{
    int i = blockIdx.x * blockDim.x + threadIdx.x;
    if (i >= n * FPAD) return;
    int r = i >> 4, c = i & 15;
    xp[i] = (c < FIN) ? x[(size_t)r * FIN + c] : 0.f;
}

// ---------------------------------------------------------------------------
// Elementwise / attention kernels
// ---------------------------------------------------------------------------
__global__ void fill_f32(float* __restrict__ p, float v, long n)
{
    long i = (long)blockIdx.x * blockDim.x + threadIdx.x;
    if (i < n) p[i] = v;
}

// One wave per node: coalesced lane-strided reads + shuffle reduction.
template<int C>
__global__ __launch_bounds__(256)
void alpha_kernel(const float* __restrict__ H, const float* __restrict__ asv,
                  const float* __restrict__ adv,
                  float* __restrict__ als, float* __restrict__ ald, int n)
{
    const int lane = threadIdx.x & 31;
    const int node = blockIdx.x * 8 + (threadIdx.x >> 5);
    if (node >= n) return;
    float s = 0.f, d = 0.f;
    #pragma unroll
    for (int c = lane; c < C; c += 32) {
        float h = H[(size_t)node * C + c];
        s = fmaf(h, asv[c], s);
        d = fmaf(h, adv[c], d);
    }
    #pragma unroll
    for (int off = 16; off > 0; off >>= 1) {
        s += __shfl_xor(s, off, 32);
        d += __shfl_xor(d, off, 32);
    }
    if (lane == 0) { als[node] = s; ald[node] = d; }
}

__device__ __forceinline__ void atomicMaxF32(float* addr, float val)
{
    // Sign-split trick; memory initialized to -inf.
    if (val >= 0.f) atomicMax((int*)addr, __float_as_int(val));
    else            atomicMin((unsigned int*)addr, __float_as_uint(val));
}

__global__ void edge_max(const int* __restrict__ srcE, const int* __restrict__ dstE,
                         const float* __restrict__ als, const float* __restrict__ ald,
                         float* __restrict__ m, float* __restrict__ ebuf, int E, int n)
{
    int i = blockIdx.x * blockDim.x + threadIdx.x;
    if (i >= E + n) return;
    int s, d;
    if (i < E) { s = srcE[i]; d = dstE[i]; } else { s = d = i - E; }   // self-loops
    float e = als[s] + ald[d];
    e = e > 0.f ? e : NEG_SLOPE * e;            // leaky_relu
    ebuf[i] = e;
    atomicMaxF32(&m[d], e);
}

__global__ void edge_exp(const int* __restrict__ dstE, const float* __restrict__ m,
                         float* __restrict__ ebuf, float* __restrict__ den, int E, int n)
{
    int i = blockIdx.x * blockDim.x + threadIdx.x;
    if (i >= E + n) return;
    int d = (i < E) ? dstE[i] : (i - E);
    float t = __expf(ebuf[i] - m[d]);
    ebuf[i] = t;
    atomicAdd(&den[d], t);
}

template<int C>
__global__ void edge_aggregate(const int* __restrict__ srcE, const int* __restrict__ dstE,
                               const float* __restrict__ ebuf, const float* __restrict__ den,
                               const float* __restrict__ H, float* __restrict__ acc, int E, int n)
{
    int i = blockIdx.x * blockDim.x + threadIdx.x;
    if (i >= E + n) return;
    int s, d;
    if (i < E) { s = srcE[i]; d = dstE[i]; } else { s = d = i - E; }
    const int   c0 = blockIdx.y * 4;
    const float a  = ebuf[i] / den[d];
    const float4 hv = *(const float4*)(H + (size_t)s * C + c0);
    float* dp = acc + (size_t)d * C + c0;
    atomicAdd(dp + 0, a * hv.x);
    atomicAdd(dp + 1, a * hv.y);
    atomicAdd(dp + 2, a * hv.z);
    atomicAdd(dp + 3, a * hv.w);
}

template<int C>
__global__ void bias_relu(float* __restrict__ acc, const float* __restrict__ bias, int n)
{
    long i = (long)blockIdx.x * blockDim.x + threadIdx.x;
    if (i >= (long)n * C) return;
    int c = (int)(i & (C - 1));                 // C is a power of two
    float v = acc[i] + bias[c];
    acc[i] = v > 0.f ? v : 0.f;
}

__global__ void pool_accum(const float* __restrict__ H, const int* __restrict__ batch,
                           float* __restrict__ psum, float* __restrict__ pcnt, int n)
{
    int i = blockIdx.x * blockDim.x + threadIdx.x;
    if (i >= n) return;
    int g = batch[i];
    atomicAdd(&pcnt[g], 1.f);
    #pragma unroll
    for (int c = 0; c < NCLS; ++c)
        atomicAdd(&psum[g * NCLS + c], H[(size_t)i * NCLS + c]);
}

__global__ void pool_final(const float* __restrict__ psum, const float* __restrict__ pcnt,
                           float* __restrict__ out)
{
    int i = blockIdx.x * blockDim.x + threadIdx.x;
    if (i >= NG * NCLS) return;
    float c = pcnt[i / NCLS];
    c = c > 1.f ? c : 1.f;
    float v = psum[i] / c;
    out[i] = 1.f / (1.f + __expf(-v));
}

// ---------------------------------------------------------------------------
// Host-side orchestration
// ---------------------------------------------------------------------------
static inline int cdiv(long a, long b) { return (int)((a + b - 1) / b); }

template<int COUT>
static void attention_layer(const float* H, const float* asv, const float* adv,
                            const float* bias, const int* srcE, const int* dstE,
                            float* als, float* ald, float* mArr, float* den,
                            float* ebuf, float* acc_out, hipStream_t stream)
{
    const int Et = NE + NN;
    alpha_kernel<COUT><<<cdiv(NN, 8), 256, 0, stream>>>(H, asv, adv, als, ald, NN);
    fill_f32<<<cdiv(NN, 256), 256, 0, stream>>>(mArr, -INFINITY, NN);
    fill_f32<<<cdiv(NN, 256), 256, 0, stream>>>(den, 0.f, NN);
    fill_f32<<<cdiv((long)NN * COUT, 256), 256, 0, stream>>>(acc_out, 0.f, (long)NN * COUT);
    edge_max<<<cdiv(Et, 256), 256, 0, stream>>>(srcE, dstE, als, ald, mArr, ebuf, NE, NN);
    edge_exp<<<cdiv(Et, 256), 256, 0, stream>>>(dstE, mArr, ebuf, den, NE, NN);
    dim3 agg_grid(cdiv(Et, 256), COUT / 4);
    edge_aggregate<COUT><<<agg_grid, 256, 0, stream>>>(srcE, dstE, ebuf, den, H, acc_out, NE, NN);
    bias_relu<COUT><<<cdiv((long)NN * COUT, 256), 256, 0, stream>>>(acc_out, bias, NN);
}

extern "C" void kernel_launch(void* const* d_in, const int* in_sizes, int n_in,
                              void* d_out, int out_size, void* d_ws, size_t ws_size,
                              hipStream_t stream)
{
    (void)in_sizes; (void)n_in; (void)out_size; (void)ws_size;
    const float* x    = (const float*)d_in[0];
    const int*   ei   = (const int*)  d_in[1];
    /* edge_attr d_in[2] is unused by the reference */
    const int*   batch= (const int*)  d_in[3];
    const float* W1 = (const float*)d_in[4],  *as1 = (const float*)d_in[5];
    const float* ad1= (const float*)d_in[6],  *b1  = (const float*)d_in[7];
    const float* W2 = (const float*)d_in[8],  *as2 = (const float*)d_in[9];
    const float* ad2= (const float*)d_in[10], *b2  = (const float*)d_in[11];
    const float* W3 = (const float*)d_in[12], *as3 = (const float*)d_in[13];
    const float* ad3= (const float*)d_in[14], *b3  = (const float*)d_in[15];
    const int* srcE = ei;
    const int* dstE = ei + NE;
    float* out = (float*)d_out;

    // Workspace carve-out (~59 MB)
    float* p    = (float*)d_ws;
    float* bufA = p; p += (size_t)NN * HID;      // gemm output h
    float* bufB = p; p += (size_t)NN * HID;      // aggregation acc / next input
    float* xp   = p; p += (size_t)NN * FPAD;     // padded input features
    float* als  = p; p += NN;
    float* ald  = p; p += NN;
    float* mArr = p; p += NN;
    float* den  = p; p += NN;
    float* ebuf = p; p += (NE + NN);
    float* psum = p; p += NG * NCLS;
    float* pcnt = p; p += NG;

    // ---- Layer 1: 11 (padded to 16) -> 128 ----   (NN % 16 == 0 -> no guard)
    pack_x<<<cdiv((long)NN * FPAD, 256), 256, 0, stream>>>(x, xp, NN);
    gat_gemm_wmma<FPAD, HID, false><<<cdiv(NN, 16), 256, 0, stream>>>(xp, W1, bufA, NN);
    attention_layer<HID>(bufA, as1, ad1, b1, srcE, dstE, als, ald, mArr, den, ebuf, bufB, stream);

    // ---- Layer 2: 128 -> 128 ----
    gat_gemm_wmma<HID, HID, false><<<cdiv(NN, 16), 256, 0, stream>>>(bufB, W2, bufA, NN);
    attention_layer<HID>(bufA, as2, ad2, b2, srcE, dstE, als, ald, mArr, den, ebuf, bufB, stream);

    // ---- Layer 3: 128 -> 16 ----   (128-row blocks -> tail guard)
    gat_gemm_wmma<HID, NCLS, true><<<cdiv(NN, 128), 256, 0, stream>>>(bufB, W3, bufA, NN);
    attention_layer<NCLS>(bufA, as3, ad3, b3, srcE, dstE, als, ald, mArr, den, ebuf, bufB, stream);

    // ---- Global mean pool + sigmoid ----
    fill_f32<<<cdiv(NG * NCLS + NG, 256), 256, 0, stream>>>(psum, 0.f, NG * NCLS + NG);
    pool_accum<<<cdiv(NN, 256), 256, 0, stream>>>(bufB, batch, psum, pcnt, NN);
    pool_final<<<cdiv(NG * NCLS, 256), 256, 0, stream>>>(psum, pcnt, out);
}